// MultiLoss_15333033247306
// MI455X (gfx1250) — compile-verified
//
#include <hip/hip_runtime.h>

#define OFFS 1e-6f
#define EPSV 1e-6f

#define FEAT_G 16   // floats per box record in global workspace (64B -> b128 friendly)
#define FEAT_L 20   // floats per box record in LDS (80B: 16B-aligned, bank-conflict free)
#define TM 64
#define TN 64

#if __has_builtin(__builtin_amdgcn_global_load_async_to_lds_b128) && \
    __has_builtin(__builtin_amdgcn_s_wait_asynccnt)
#define USE_ASYNC_LDS 1
#else
#define USE_ASYNC_LDS 0
#endif

#if USE_ASYNC_LDS
typedef int v4i_cdna5 __attribute__((ext_vector_type(4)));
typedef __attribute__((address_space(1))) v4i_cdna5* gpu_gptr_b128;
typedef __attribute__((address_space(3))) v4i_cdna5* gpu_lptr_b128;

__device__ __forceinline__ void async_copy_b128(const float* g, float* l) {
    // CDNA5 async path: tracked by ASYNCcnt, waited via s_wait_asynccnt
    __builtin_amdgcn_global_load_async_to_lds_b128(
        (gpu_gptr_b128)g, (gpu_lptr_b128)l, 0, 0);
}
#endif

typedef float v4f_cdna5 __attribute__((ext_vector_type(4)));

// Non-temporal b128 store: output is 201MB of write-once data; TH=NT keeps it
// from thrashing the 192MB L2 (outputs are never re-read).
__device__ __forceinline__ void store_nt4(float* p, float a, float b, float c, float d) {
    v4f_cdna5 v = {a, b, c, d};
#if __has_builtin(__builtin_nontemporal_store)
    __builtin_nontemporal_store(v, (v4f_cdna5*)p);
#else
    *(v4f_cdna5*)p = v;
#endif
}

__device__ __forceinline__ float fast_div(float a, float b) {
#if __has_builtin(__builtin_amdgcn_rcpf)
    float r = __builtin_amdgcn_rcpf(b);
    r = fmaf(fmaf(-b, r, 1.0f), r, r);   // one Newton step -> ~fp32 accurate
    return a * r;
#else
    return a / b;
#endif
}

// Per-box features, packed into 16 floats:
// [0..3]  bev standup  (xmin,ymin,xmax,ymax)   -- rotated by yaw
// [4..7]  face standup (xmin,ymin,xmax,ymax)   -- axis aligned (y,z) x (l,h)
// [8..12] zmin, zmax, vol, area_bev, area_face
__global__ void preprocess_kernel(const float* __restrict__ boxes,
                                  float* __restrict__ feat, int n) {
    int i = blockIdx.x * blockDim.x + threadIdx.x;
    if (i >= n) return;
    const float* b = boxes + (size_t)i * 7;
    float x = b[0], y = b[1], z = b[2];
    float w = b[3], l = b[4], h = b[5], yaw = b[6];

    float c = cosf(yaw), s = sinf(yaw);
    float hw = 0.5f * w, hl = 0.5f * l, hh = 0.5f * h;
    // rotated corner: X = cx*c + cy*s ; Y = -cx*s + cy*c  with cx=+-hw, cy=+-hl
    float ex = fabsf(hw * c) + fabsf(hl * s);
    float ey = fabsf(hw * s) + fabsf(hl * c);
    float bx0 = x - ex, bx1 = x + ex;
    float by0 = y - ey, by1 = y + ey;
    float area_b = (bx1 - bx0 + OFFS) * (by1 - by0 + OFFS);

    float fx0 = y - hl, fx1 = y + hl;
    float fy0 = z - hh, fy1 = z + hh;
    float area_f = (fx1 - fx0 + OFFS) * (fy1 - fy0 + OFFS);

    float4* o = (float4*)(feat + (size_t)i * FEAT_G);
    o[0] = make_float4(bx0, by0, bx1, by1);
    o[1] = make_float4(fx0, fy0, fx1, fy1);
    o[2] = make_float4(z - h, z, w * l * h, area_b);
    o[3] = make_float4(area_f, 0.0f, 0.0f, 0.0f);
}

struct BoxF {
    float bx0, by0, bx1, by1;
    float fx0, fy0, fx1, fy1;
    float zmin, zmax, vol, areaB, areaF;
};

__device__ __forceinline__ BoxF load_feat(const float* p) {
    BoxF f;
    f.bx0 = p[0];  f.by0 = p[1];  f.bx1 = p[2];  f.by1 = p[3];
    f.fx0 = p[4];  f.fy0 = p[5];  f.fx1 = p[6];  f.fy1 = p[7];
    f.zmin = p[8]; f.zmax = p[9]; f.vol = p[10];
    f.areaB = p[11]; f.areaF = p[12];
    return f;
}

__global__ __launch_bounds__(256)
void iou_tile_kernel(const float* __restrict__ featP, const float* __restrict__ featT,
                     float* __restrict__ out, int nP, int nT) {
    __shared__ float sP[TM * FEAT_L];
    __shared__ float sT[TN * FEAT_L];

    const int tid   = threadIdx.x;
    const int baseM = blockIdx.y * TM;
    const int baseN = blockIdx.x * TN;
    const size_t NN = (size_t)nP * (size_t)nT;

    // ---- stage both feature tiles into LDS (256 threads x one b128 each side)
    {
        const int row  = tid >> 2;                 // 0..63
        const int q    = tid & 3;                  // which float4 of the record
        const int srcP = min(baseM + row, nP - 1);
        const int srcT = min(baseN + row, nT - 1);
        const float* gp = featP + (size_t)srcP * FEAT_G + q * 4;
        const float* gt = featT + (size_t)srcT * FEAT_G + q * 4;
        float* lp = &sP[row * FEAT_L + q * 4];
        float* lt = &sT[row * FEAT_L + q * 4];

        __builtin_prefetch(gp, 0, 0);              // probe-confirmed: global_prefetch_b8

#if USE_ASYNC_LDS
        async_copy_b128(gp, lp);
        async_copy_b128(gt, lt);
        __builtin_amdgcn_s_wait_asynccnt(0);
#else
        *(float4*)lp = *(const float4*)gp;
        *(float4*)lt = *(const float4*)gt;
#endif
    }
    __syncthreads();

    // ---- compute 4x4 micro-tile per thread
    const int n4 = tid & 15;        // column group: 4 consecutive targets
    const int mg = tid >> 4;        // row phase: rows mg, mg+16, mg+32, mg+48

    BoxF t[4];
#pragma unroll
    for (int j = 0; j < 4; ++j)
        t[j] = load_feat(&sT[(n4 * 4 + j) * FEAT_L]);

#pragma unroll
    for (int i = 0; i < 4; ++i) {
        const int mL = mg + 16 * i;
        const BoxF p = load_feat(&sP[mL * FEAT_L]);

        float ob[4], og[4], o3[4];
#pragma unroll
        for (int j = 0; j < 4; ++j) {
            const BoxF q = t[j];
            // BEV overlap / IoU
            float ltx = fmaxf(p.bx0, q.bx0), lty = fmaxf(p.by0, q.by0);
            float rbx = fminf(p.bx1, q.bx1), rby = fminf(p.by1, q.by1);
            float bw  = fmaxf(rbx - ltx + OFFS, 0.0f);
            float bh  = fmaxf(rby - lty + OFFS, 0.0f);
            float ovb = bw * bh;
            float ib  = fast_div(ovb, p.areaB + q.areaB - ovb);
            ob[j] = fminf(fmaxf(ib, EPSV), 1.0f - EPSV);
            // Face overlap / IoU
            float gltx = fmaxf(p.fx0, q.fx0), glty = fmaxf(p.fy0, q.fy0);
            float grbx = fminf(p.fx1, q.fx1), grby = fminf(p.fy1, q.fy1);
            float gw   = fmaxf(grbx - gltx + OFFS, 0.0f);
            float gh   = fmaxf(grby - glty + OFFS, 0.0f);
            float ovg  = gw * gh;
            float ig   = fast_div(ovg, p.areaF + q.areaF - ovg);
            og[j] = fminf(fmaxf(ig, EPSV), 1.0f - EPSV);
            // 3D IoU
            float ovh = fmaxf(fminf(p.zmax, q.zmax) - fmaxf(p.zmin, q.zmin), 0.0f);
            float ov3 = ovb * ovh;
            float den = fmaxf(p.vol + q.vol - ov3, EPSV);
            float i3  = fast_div(ov3, den);
            o3[j] = fminf(fmaxf(i3, EPSV), 1.0f);
        }

        const int mG = baseM + mL;
        const int nG = baseN + n4 * 4;
        if (mG < nP) {
            size_t off = (size_t)mG * (size_t)nT + (size_t)nG;
            if (((nT & 3) == 0) && (nG + 3 < nT)) {
                store_nt4(out + off,          ob[0], ob[1], ob[2], ob[3]);
                store_nt4(out + NN + off,     og[0], og[1], og[2], og[3]);
                store_nt4(out + 2 * NN + off, o3[0], o3[1], o3[2], o3[3]);
            } else {
#pragma unroll
                for (int j = 0; j < 4; ++j) {
                    if (nG + j < nT) {
                        out[off + j]          = ob[j];
                        out[NN + off + j]     = og[j];
                        out[2 * NN + off + j] = o3[j];
                    }
                }
            }
        }
    }
}

extern "C" void kernel_launch(void* const* d_in, const int* in_sizes, int n_in,
                              void* d_out, int out_size, void* d_ws, size_t ws_size,
                              hipStream_t stream) {
    const float* pred   = (const float*)d_in[0];
    const float* target = (const float*)d_in[1];
    const int nP = in_sizes[0] / 7;
    const int nT = in_sizes[1] / 7;

    float* featP = (float*)d_ws;
    float* featT = featP + (size_t)nP * FEAT_G;

    preprocess_kernel<<<(nP + 255) / 256, 256, 0, stream>>>(pred, featP, nP);
    preprocess_kernel<<<(nT + 255) / 256, 256, 0, stream>>>(target, featT, nT);

    dim3 grid((nT + TN - 1) / TN, (nP + TM - 1) / TM);
    iou_tile_kernel<<<grid, 256, 0, stream>>>(featP, featT, (float*)d_out, nP, nT);
}